// LlamaAttention_51745765982929
// MI455X (gfx1250) — compile-verified
//
#include <hip/hip_runtime.h>
#include <hip/hip_bf16.h>

#define BB   32
#define SS   4096
#define HH   32
#define HKV  8
#define DD   128
#define HID  4096
#define QKVLD 6144   // (H + 2*HKV) * D

typedef __attribute__((ext_vector_type(2))) float v2f;
typedef __attribute__((ext_vector_type(8))) float v8f;

// ---------------------------------------------------------------------------
// Common WMMA GEMM: C(32 x 128) = A(32 x 4096) * Bmat(4096 x ldb)[:, col0:col0+128]
// 8 waves (256 threads); wave w owns 16 output columns; A staged in LDS in
// 32x64 K-panels; result left in ldsC (32 rows x 128 cols).
// ---------------------------------------------------------------------------
__device__ __forceinline__ void gemm_32x128(const float* __restrict__ A,
                                            const float* __restrict__ Bmat,
                                            int ldb, int col0,
                                            float* ldsA, float* ldsC) {
  const int tid  = threadIdx.x;
  const int lane = tid & 31;
  const int wv   = tid >> 5;
  const int m0   = lane & 15;        // A row (low tile) / B column-in-tile
  const int hi   = lane >> 4;
  const int kb   = hi << 1;          // K pair base {0,2}
  const long ncol = (long)col0 + wv * 16 + m0;

  v8f c0 = (v8f)0.f;                 // rows 0..15
  v8f c1 = (v8f)0.f;                 // rows 16..31

  for (int k0 = 0; k0 < HID; k0 += 64) {
    // stage A panel: 32 rows x 64 k (2048 floats, 8 per thread, coalesced)
#pragma unroll
    for (int i = 0; i < 8; ++i) {
      int e = tid + i * 256;
      int r = e >> 6, kk = e & 63;
      ldsA[e] = A[(long)r * HID + k0 + kk];
    }
    __syncthreads();
#pragma unroll
    for (int kk = 0; kk < 64; kk += 4) {
      v2f a0, a1, bb;
      a0.x = ldsA[m0 * 64 + kk + kb];
      a0.y = ldsA[m0 * 64 + kk + kb + 1];
      a1.x = ldsA[(m0 + 16) * 64 + kk + kb];
      a1.y = ldsA[(m0 + 16) * 64 + kk + kb + 1];
      bb.x = Bmat[(long)(k0 + kk + kb) * ldb + ncol];
      bb.y = Bmat[(long)(k0 + kk + kb + 1) * ldb + ncol];
      c0 = __builtin_amdgcn_wmma_f32_16x16x4_f32(false, a0, false, bb, (short)0, c0, false, false);
      c1 = __builtin_amdgcn_wmma_f32_16x16x4_f32(false, a1, false, bb, (short)0, c1, false, false);
    }
    __syncthreads();
  }
  // spill C tiles to LDS (ISA C layout: vgpr i -> row i + hi*8, col = lane&15)
#pragma unroll
  for (int i = 0; i < 8; ++i) {
    int r = i + (hi << 3);
    ldsC[r * 128 + wv * 16 + m0]        = c0[i];
    ldsC[(16 + r) * 128 + wv * 16 + m0] = c1[i];
  }
  __syncthreads();
}

// ---------------------------------------------------------------------------
// Kernel 0: zero the "OOB source" block used by async loads in attention.
// ---------------------------------------------------------------------------
__global__ __launch_bounds__(128)
void zero_ws_kernel(float* __restrict__ zero_ws) {
  zero_ws[threadIdx.x] = 0.f;
}

// ---------------------------------------------------------------------------
// Kernel 1: QKV projection (WMMA) + RoPE; writes q/k_new/v_new to workspace.
// grid.x = 48 heads (0..31 q | 32..39 k | 40..47 v), block = 256.
// ---------------------------------------------------------------------------
__global__ __launch_bounds__(256)
void qkv_rope_kernel(const float* __restrict__ hidden,
                     const float* __restrict__ w_qkv,
                     const float* __restrict__ cosb,
                     const float* __restrict__ sinb,
                     float* __restrict__ q_ws,
                     float* __restrict__ k_ws,
                     float* __restrict__ v_ws) {
  __shared__ float ldsA[32 * 64];
  __shared__ float ldsC[32 * 128];
  const int head = blockIdx.x;
  gemm_32x128(hidden, w_qkv, QKVLD, head * DD, ldsA, ldsC);

  for (int e = threadIdx.x; e < 32 * 128; e += 256) {
    int b = e >> 7, d = e & 127;
    float x = ldsC[e];
    float outv;
    if (head < HH + HKV) { // q or k head: apply RoPE
      float rot = (d < 64) ? -ldsC[(b << 7) + d + 64] : ldsC[(b << 7) + d - 64];
      outv = x * cosb[(b << 7) + d] + rot * sinb[(b << 7) + d];
    } else {
      outv = x;
    }
    if (head < HH)            q_ws[((long)b * HH + head) * DD + d] = outv;
    else if (head < HH + HKV) k_ws[((long)b * HKV + (head - HH)) * DD + d] = outv;
    else                      v_ws[((long)b * HKV + (head - HH - HKV)) * DD + d] = outv;
  }
}

// ---------------------------------------------------------------------------
// Kernel 2: flash attention, one wave per (b, kv head). K/V tiles streamed
// with global_load_async_to_lds_b128 (ASYNCcnt) into double-buffered LDS;
// Q*K^T and P*V via f32 WMMA. New token's K/V substituted at s == cache_len.
// grid.x = B*HKV = 256, block = 32 (wave32).
// ---------------------------------------------------------------------------
__global__ __launch_bounds__(32)
void attn_kernel(const float* __restrict__ q_ws,
                 const float* __restrict__ k_ws,
                 const float* __restrict__ v_ws,
                 const float* __restrict__ k_cache,
                 const float* __restrict__ v_cache,
                 const int* __restrict__ cache_lens,
                 const float* __restrict__ zero_ws,
                 float* __restrict__ attn_ws) {
  __shared__ float qs[4 * 128];
  __shared__ float kt[2][16 * 128];
  __shared__ float vt[2][16 * 128];
  __shared__ float pr[4 * 16];

  const int b    = blockIdx.x >> 3;
  const int kv   = blockIdx.x & 7;
  const int lane = threadIdx.x;
  const int m0   = lane & 15;
  const int hi   = lane >> 4;
  const int kb   = hi << 1;
  const int len  = cache_lens[b];

  // Issue one 16x128 K tile + V tile into LDS buffer p via async copies.
  // 32 b128 chunks per row, 16 rows => 512 chunks per matrix; each lane
  // issues 16 async ops per matrix (32 total per tile).
  auto issue_tile = [&](int p, int s0t) {
#pragma unroll
    for (int it = 0; it < 16; ++it) {
      int c   = lane + it * 32;
      int row = c >> 5;
      int off = (c & 31) << 2;      // float offset within row, 16B aligned
      int sg  = s0t + row;
      const float *ksrc, *vsrc;
      if (sg < len) {
        long base = (((long)b * SS + sg) * HKV + kv) * DD + off;
        ksrc = k_cache + base;
        vsrc = v_cache + base;
      } else if (sg == len) {
        long base = ((long)b * HKV + kv) * DD + off;
        ksrc = k_ws + base;
        vsrc = v_ws + base;
      } else {
        ksrc = zero_ws + off;       // zeroed block
        vsrc = zero_ws + off;
      }
      unsigned kl = (unsigned)(unsigned long)(&kt[p][row * 128 + off]);
      unsigned vl = (unsigned)(unsigned long)(&vt[p][row * 128 + off]);
      asm volatile("global_load_async_to_lds_b128 %0, %1, off"
                   :: "v"(kl), "v"(ksrc) : "memory");
      asm volatile("global_load_async_to_lds_b128 %0, %1, off"
                   :: "v"(vl), "v"(vsrc) : "memory");
    }
  };

  // stage Q (pre-scaled by 1/sqrt(D))
  const float qscale = 0.08838834764831845f;
  for (int e = lane; e < 512; e += 32) {
    int g = e >> 7, d = e & 127;
    qs[e] = q_ws[((long)b * HH + kv * 4 + g) * DD + d] * qscale;
  }

  float mrow[4] = {-1e30f, -1e30f, -1e30f, -1e30f};
  float lsum[4] = {0.f, 0.f, 0.f, 0.f};
  v8f o[8];
#pragma unroll
  for (int t = 0; t < 8; ++t) o[t] = (v8f)0.f;

  issue_tile(0, 0);                 // prologue: tile 0 -> buffer 0

  int t = 0;
  for (int s0 = 0; s0 <= len; s0 += 16, ++t) {
    const int p = t & 1;
    const bool more = (s0 + 16) <= len;
    if (more) {
      // warm L2 two tiles ahead while we're at it
      int spf = s0 + 32 + m0; if (spf > len) spf = len;
      long pfb = (((long)b * SS + spf) * HKV + kv) * DD;
      __builtin_prefetch(k_cache + pfb, 0, 1);
      __builtin_prefetch(v_cache + pfb, 0, 1);
      issue_tile(p ^ 1, s0 + 16);   // overlap next tile's copies with compute
      asm volatile("s_wait_asynccnt 32" ::: "memory");  // tile t complete
    } else {
      asm volatile("s_wait_asynccnt 0" ::: "memory");
    }

    // scores tile: S(4x16) = Q(4x128) * K^T, as 32 WMMA f32 16x16x4 steps
    v8f sc = (v8f)0.f;
#pragma unroll
    for (int dk = 0; dk < 128; dk += 4) {
      v2f a, bb;
      a.x = (m0 < 4) ? qs[m0 * 128 + dk + kb] : 0.f;
      a.y = (m0 < 4) ? qs[m0 * 128 + dk + kb + 1] : 0.f;
      bb.x = kt[p][m0 * 128 + dk + kb];   // B[k][n] = K[s0+n][dk+k]
      bb.y = kt[p][m0 * 128 + dk + kb + 1];
      sc = __builtin_amdgcn_wmma_f32_16x16x4_f32(false, a, false, bb, (short)0, sc, false, false);
    }

    // online softmax over rows 0..3 (columns striped over lanes 0..15)
    const bool valid = (s0 + m0) <= len;
#pragma unroll
    for (int i = 0; i < 4; ++i) {
      float s = valid ? sc[i] : -3e38f;
      float tm = s;
      tm = fmaxf(tm, __shfl_xor(tm, 1));
      tm = fmaxf(tm, __shfl_xor(tm, 2));
      tm = fmaxf(tm, __shfl_xor(tm, 4));
      tm = fmaxf(tm, __shfl_xor(tm, 8));
      float nm  = fmaxf(mrow[i], tm);
      float rsc = __expf(mrow[i] - nm);
      float pp  = __expf(s - nm);
      float ps  = pp;
      ps += __shfl_xor(ps, 1);
      ps += __shfl_xor(ps, 2);
      ps += __shfl_xor(ps, 4);
      ps += __shfl_xor(ps, 8);
      lsum[i] = lsum[i] * rsc + ps;
      mrow[i] = nm;
#pragma unroll
      for (int tt = 0; tt < 8; ++tt) o[tt][i] *= rsc;
      if (hi == 0) pr[i * 16 + m0] = pp;   // same-wave LDS ops are in-order
    }

    // O(4x128) += P(4x16) * V(16x128): 4 K-steps x 8 N-tiles of WMMA
#pragma unroll
    for (int kk = 0; kk < 16; kk += 4) {
      v2f a;
      a.x = (m0 < 4) ? pr[m0 * 16 + kk + kb] : 0.f;
      a.y = (m0 < 4) ? pr[m0 * 16 + kk + kb + 1] : 0.f;
#pragma unroll
      for (int tt = 0; tt < 8; ++tt) {
        v2f bb;
        bb.x = vt[p][(kk + kb) * 128 + tt * 16 + m0];
        bb.y = vt[p][(kk + kb + 1) * 128 + tt * 16 + m0];
        o[tt] = __builtin_amdgcn_wmma_f32_16x16x4_f32(false, a, false, bb, (short)0, o[tt], false, false);
      }
    }
  }

  // normalize + write attn rows (rows 0..3 live in vgprs 0..3 of lanes 0..15)
  if (hi == 0) {
#pragma unroll
    for (int i = 0; i < 4; ++i) {
      float inv = 1.0f / lsum[i];
#pragma unroll
      for (int tt = 0; tt < 8; ++tt)
        attn_ws[(long)b * (HH * DD) + (kv * 4 + i) * DD + tt * 16 + m0] = o[tt][i] * inv;
    }
  }
}

// ---------------------------------------------------------------------------
// Kernel 3: output projection out(32x4096) = attn(32x4096) @ w_o(4096x4096)
// grid.x = 32 (128 cols each), block = 256.
// ---------------------------------------------------------------------------
__global__ __launch_bounds__(256)
void out_proj_kernel(const float* __restrict__ attn_ws,
                     const float* __restrict__ w_o,
                     float* __restrict__ out) {
  __shared__ float ldsA[32 * 64];
  __shared__ float ldsC[32 * 128];
  const int col0 = blockIdx.x * 128;
  gemm_32x128(attn_ws, w_o, HID, col0, ldsA, ldsC);
  for (int e = threadIdx.x; e < 32 * 128; e += 256) {
    int b = e >> 7, d = e & 127;
    out[(long)b * HID + col0 + d] = ldsC[e];
  }
}

extern "C" void kernel_launch(void* const* d_in, const int* in_sizes, int n_in,
                              void* d_out, int out_size, void* d_ws, size_t ws_size,
                              hipStream_t stream) {
  const float* hidden     = (const float*)d_in[0];
  const float* w_qkv      = (const float*)d_in[1];
  const float* w_o        = (const float*)d_in[2];
  const float* cosb       = (const float*)d_in[3];
  const float* sinb       = (const float*)d_in[4];
  const float* k_cache    = (const float*)d_in[5];
  const float* v_cache    = (const float*)d_in[6];
  const int*   cache_lens = (const int*)d_in[7];

  float* q_ws    = (float*)d_ws;                 // B*H*D   = 131072 floats
  float* k_ws    = q_ws + (long)BB * HH * DD;    // B*HKV*D =  32768 floats
  float* v_ws    = k_ws + (long)BB * HKV * DD;
  float* attn_ws = v_ws + (long)BB * HKV * DD;   // B*H*D   = 131072 floats
  float* zero_ws = attn_ws + (long)BB * HH * DD; // 128 floats, zero-filled

  zero_ws_kernel<<<1, 128, 0, stream>>>(zero_ws);
  qkv_rope_kernel<<<HH + 2 * HKV, 256, 0, stream>>>(hidden, w_qkv, cosb, sinb,
                                                    q_ws, k_ws, v_ws);
  attn_kernel<<<BB * HKV, 32, 0, stream>>>(q_ws, k_ws, v_ws, k_cache, v_cache,
                                           cache_lens, zero_ws, attn_ws);
  out_proj_kernel<<<HID / 128, 256, 0, stream>>>(attn_ws, w_o, (float*)d_out);
}